// SNN_86096914416141
// MI455X (gfx1250) — compile-verified
//
#include <hip/hip_runtime.h>
#include <stdint.h>

// SNN adaptive-LIF scan, [B=64, N=1024, T=512] float32.
// Bandwidth-bound (268 MB @ 23.3 TB/s ~ 11.5 us floor). No matmul -> no WMMA.
// CDNA5 path: async global->LDS loads (ASYNCcnt) + s_wait_asynccnt,
// double-buffered LDS transpose staging, NT hints on the one-shot streams.

#define SNN_T    512                 // timesteps
#define SNN_ROWS 256                 // (b,n) pairs per workgroup == blockDim.x
#define SNN_TC   16                  // timesteps per LDS chunk
#define SNN_STR  20                  // row pitch in floats (TC+4): 80B, 16B-aligned, bank-skewed
#define SNN_NCH  (SNN_T / SNN_TC)    // 32 chunks

typedef __attribute__((ext_vector_type(4))) float v4f;  // native clang vector (NT-builtin OK)

__device__ __forceinline__ float snn_step(float xv, float beta_c, float p_c, float b_c,
                                          float& mem, float& refac, float& a,
                                          float& vth, float& s) {
  // spike-triggered refractory reset (uses previous step's spike)
  refac = (s > 0.0f) ? 0.0f : refac;
  float ic = (refac < 2.0f) ? 0.0f : xv;     // mask input during refractory window
  refac = __fadd_rn(refac, 1.0f);
  float nm = __fadd_rn(__fmul_rn(mem, beta_c), ic);   // leaky integration
  float u  = __fsub_rn(nm, vth);
  s = (u > 0.0f) ? 1.0f : 0.0f;              // fast-sigmoid fwd = heaviside
  mem = __fmul_rn(nm, __fsub_rn(1.0f, s));   // reset-to-zero on spike
  a   = __fadd_rn(__fmul_rn(p_c, a), s);     // adaptation state
  vth = __fadd_rn(__fmul_rn(b_c, a), 1.0f);  // adaptive threshold
  return s;
}

__global__ __launch_bounds__(SNN_ROWS) void SNN_86096914416141_kernel(
    const float* __restrict__ x, const float* __restrict__ beta,
    const float* __restrict__ p, const float* __restrict__ b,
    float* __restrict__ out, int Nn) {
  // 2 input buffers (double-buffered async loads) + 1 output staging buffer.
  // 60 KB total -> ~5 workgroups (40 waves) per 320KB WGP.
  __shared__ __align__(16) float in_lds[2][SNN_ROWS * SNN_STR];
  __shared__ __align__(16) float out_lds[SNN_ROWS * SNN_STR];

  const int tid = threadIdx.x;
  const unsigned pairBase32 = (unsigned)blockIdx.x * SNN_ROWS;
  const size_t pairBase = (size_t)pairBase32;

  // per-neuron clamped parameters; N is a power of two (1024) -> mask, no idiv
  const int n = (int)((pairBase32 + (unsigned)tid) & (unsigned)(Nn - 1));
  const float beta_c = fminf(fmaxf(beta[n], 0.001f), 0.999f);
  const float p_c    = fminf(fabsf(p[n]), 0.999f);
  const float b_c    = fminf(fmaxf(fabsf(b[n]), 0.001f), 1.0f);

  // coalesced tile mapping: 4 lanes cover one row's 64B chunk; a wave covers
  // 8 rows x 64B contiguous runs; 4 passes cover all 256 rows.
  const int c4   = tid & 3;   // float4 column within the row chunk
  const int rsub = tid >> 2;  // 0..63

  // Issue async global->LDS loads for chunk `ch` into buffer `buf`.
  // 4 x b128 per thread per chunk -> ASYNCcnt += 4 per wave per chunk.
  // x is consumed exactly once -> non-temporal load hint.
  auto issue = [&](int ch, int buf) {
    const size_t t0 = (size_t)ch * SNN_TC;
#pragma unroll
    for (int pass = 0; pass < 4; ++pass) {
      const int row = pass * 64 + rsub;
      const float* g = x + (pairBase + (size_t)row) * SNN_T + t0 + 4 * c4;
      unsigned loff =
          (unsigned)(uintptr_t)&in_lds[buf][row * SNN_STR + 4 * c4];  // LDS byte offset
#if defined(__gfx1250__)
      asm volatile("global_load_async_to_lds_b128 %0, %1, off th:TH_LOAD_NT"
                   :: "v"(loff), "v"(g)
                   : "memory");
#endif
    }
  };

  // scan state (registers, persist across chunks)
  float mem = 0.0f, a = 0.0f, vth = 1.0f, refac = 2.0f, s = 0.0f;

  issue(0, 0);  // prologue: fill buffer 0

  for (int ch = 0; ch < SNN_NCH; ++ch) {
    const int buf = ch & 1;
    if (ch + 1 < SNN_NCH) {
      issue(ch + 1, buf ^ 1);  // prefetch next chunk (outstanding: 8)
#if defined(__gfx1250__)
      // async loads complete in order: <=4 left => chunk `ch` fully landed
      asm volatile("s_wait_asynccnt 0x4" ::: "memory");
#endif
    } else {
#if defined(__gfx1250__)
      asm volatile("s_wait_asynccnt 0x0" ::: "memory");
#endif
    }
    __syncthreads();  // chunk `ch` visible to all waves; fences in_lds reuse

    // ---- compute: 16 sequential steps from my row's staged time-series ----
    const v4f* xin  = (const v4f*)&in_lds[buf][tid * SNN_STR];
    v4f*       sout = (v4f*)&out_lds[tid * SNN_STR];
#pragma unroll
    for (int q = 0; q < SNN_TC / 4; ++q) {
      v4f xv = xin[q];
      v4f sv;
      sv.x = snn_step(xv.x, beta_c, p_c, b_c, mem, refac, a, vth, s);
      sv.y = snn_step(xv.y, beta_c, p_c, b_c, mem, refac, a, vth, s);
      sv.z = snn_step(xv.z, beta_c, p_c, b_c, mem, refac, a, vth, s);
      sv.w = snn_step(xv.w, beta_c, p_c, b_c, mem, refac, a, vth, s);
      sout[q] = sv;
    }
    __syncthreads();  // out_lds RAW fence (includes DScnt flush)

    // ---- coalesced non-temporal writeback of staged spikes ----
    const size_t t0 = (size_t)ch * SNN_TC;
#pragma unroll
    for (int pass = 0; pass < 4; ++pass) {
      const int row = pass * 64 + rsub;
      v4f v = *(const v4f*)&out_lds[row * SNN_STR + 4 * c4];
      __builtin_nontemporal_store(
          v, (v4f*)&out[(pairBase + (size_t)row) * SNN_T + t0 + 4 * c4]);
    }
    // next iteration's first __syncthreads doubles as the out_lds WAR fence
  }
}

extern "C" void kernel_launch(void* const* d_in, const int* in_sizes, int n_in,
                              void* d_out, int out_size, void* d_ws, size_t ws_size,
                              hipStream_t stream) {
  (void)n_in; (void)d_ws; (void)ws_size;
  const float* x    = (const float*)d_in[0];  // [B, N, T] f32
  const float* beta = (const float*)d_in[1];  // [N]
  const float* p    = (const float*)d_in[2];  // [N]
  const float* b    = (const float*)d_in[3];  // [N]
  float* out        = (float*)d_out;          // [B, N, T] f32

  const int Nn    = in_sizes[1];              // 1024 (power of two)
  const int pairs = in_sizes[0] / SNN_T;      // B*N = 65536
  const int grid  = pairs / SNN_ROWS;         // 256 workgroups
  (void)out_size;

  hipLaunchKernelGGL(SNN_86096914416141_kernel, dim3(grid), dim3(SNN_ROWS), 0,
                     stream, x, beta, p, b, out, Nn);
}